// LorentzForecaster_87668872446028
// MI455X (gfx1250) — compile-verified
//
#include <hip/hip_runtime.h>
#include <math.h>

// ---------------- model dims ----------------
#define B_      1024
#define NF_     21
#define SEG_    24
#define INP_    504           // SEG*NF
#define NSEG_   14
#define ED_     128
#define HD_     256
#define NSTEPS_ 4
#define NROWS_  (4*B_*NSEG_)  // 57344 encoder rows
#define RPE_    (B_*NSEG_)    // 14336 rows per expert
#define EPS_    1e-7f

typedef float v2f __attribute__((ext_vector_type(2)));
typedef float v8f __attribute__((ext_vector_type(8)));

// ---------------- wave helpers (wave32) ----------------
__device__ __forceinline__ float wsum(float v) {
#pragma unroll
  for (int o = 16; o > 0; o >>= 1) v += __shfl_xor(v, o, 32);
  return v;
}
__device__ __forceinline__ float bcast0(float v) { return __shfl(v, 0, 32); }

// factor g such that logmap0(projx(safe_expmap0(y))) = [0, g*y_spatial],
// given S = sum of squares of spatial part of y
__device__ __forceinline__ float chain_g(float S) {
  float n  = sqrtf(fmaxf(S, 1e-12f));
  float th = fminf(n, 10.f);
  float f  = sinhf(th) / n;
  float S2 = f * f * S;
  float q0 = sqrtf(1.f + S2);
  float d  = acoshf(fmaxf(q0, 1.f + EPS_));
  float sn = sqrtf(fmaxf(S2, 1e-12f));
  return d * f / sn;
}

// ---------------- 1) per-row stats: normalization + first lorentz chain as affine map ----------------
__global__ __launch_bounds__(256) void stats_kernel(
    const float* __restrict__ t0, const float* __restrict__ t1,
    const float* __restrict__ t2, const float* __restrict__ t3,
    const float* __restrict__ es_p,
    float* __restrict__ scaleArr, float* __restrict__ shiftArr)
{
  int wave = (blockIdx.x * blockDim.x + threadIdx.x) >> 5;
  int lane = threadIdx.x & 31;
  if (wave >= NROWS_) return;
  int e = wave / RPE_, r = wave % RPE_;
  const float* src = (e == 0 ? t0 : (e == 1 ? t1 : (e == 2 ? t2 : t3))) + (size_t)r * INP_;
  float s = 0.f, ss = 0.f;
  for (int j = lane; j < INP_; j += 32) { float v = src[j]; s += v; ss += v * v; }
  s = wsum(s); ss = wsum(ss);
  float mu  = s / (float)INP_;
  float var = fmaxf((ss - (float)INP_ * mu * mu) / (float)(INP_ - 1), 0.f);
  float sd  = sqrtf(var) + 1e-6f;
  float t   = tanhf(es_p[0]);
  float a   = t / sd;
  float x0  = a * (src[0] - mu);
  float Sall = a * a * fmaxf(ss - (float)INP_ * mu * mu, 0.f);
  float S1   = fmaxf(Sall - x0 * x0, 0.f);
  float g    = chain_g(S1);     // single chain: A0 = logmap0(projx(expmap0(xnorm)))
  if (lane == 0) { scaleArr[wave] = g * a; shiftArr[wave] = -g * a * mu; }
}

// ---------------- 2) fused layer-0 GEMM: 16x64 tile per wave, A built on the fly ----------------
__global__ __launch_bounds__(256) void gemm1_fused_kernel(
    const float* __restrict__ t0, const float* __restrict__ t1,
    const float* __restrict__ t2, const float* __restrict__ t3,
    const float* __restrict__ scaleArr, const float* __restrict__ shiftArr,
    const float* __restrict__ W, const float* __restrict__ bias,
    float* __restrict__ C)
{
  int wave = (blockIdx.x * blockDim.x + threadIdx.x) >> 5;
  int lane = threadIdx.x & 31;
  const int ngroups = ED_ / 64;            // 2 groups of 64 cols
  const int mtiles  = NROWS_ / 16;
  if (wave >= mtiles * ngroups) return;
  int mt = wave / ngroups, ng = wave % ngroups;
  int row = (mt << 4) + (lane & 15);
  int e = row / RPE_, r = row % RPE_;
  const float* src = (e == 0 ? t0 : (e == 1 ? t1 : (e == 2 ? t2 : t3))) + (size_t)r * INP_;
  float sc = scaleArr[row], sh = shiftArr[row];
  int khalf = (lane >> 4) << 1;            // 0 or 2
  int col0  = (ng << 6) + (lane & 15);     // base column for subtile 0
  const float* Wp = W + (size_t)e * INP_ * ED_;
  v8f acc0 = {}, acc1 = {}, acc2 = {}, acc3 = {};
  for (int k0 = 0; k0 < INP_; k0 += 4) {
    int ka = k0 + khalf;
    v2f a;
    a.x = (ka == 0) ? 0.f : sc * src[ka] + sh;   // component 0 of logmap0 output is 0
    a.y = sc * src[ka + 1] + sh;
    const float* wp = Wp + (size_t)ka * ED_ + col0;
    v2f b0, b1, b2, b3;
    b0.x = wp[0];  b0.y = wp[ED_];
    b1.x = wp[16]; b1.y = wp[ED_ + 16];
    b2.x = wp[32]; b2.y = wp[ED_ + 32];
    b3.x = wp[48]; b3.y = wp[ED_ + 48];
    acc0 = __builtin_amdgcn_wmma_f32_16x16x4_f32(false, a, false, b0, (short)0, acc0, false, false);
    acc1 = __builtin_amdgcn_wmma_f32_16x16x4_f32(false, a, false, b1, (short)0, acc1, false, false);
    acc2 = __builtin_amdgcn_wmma_f32_16x16x4_f32(false, a, false, b2, (short)0, acc2, false, false);
    acc3 = __builtin_amdgcn_wmma_f32_16x16x4_f32(false, a, false, b3, (short)0, acc3, false, false);
  }
  int crow = (mt << 4) + ((lane >> 4) << 3);
  union { v8f v; float f[8]; } u;
#pragma unroll
  for (int i = 0; i < 4; ++i) {
    u.v = (i == 0) ? acc0 : (i == 1) ? acc1 : (i == 2) ? acc2 : acc3;
    int ccol = (ng << 6) + (i << 4) + (lane & 15);
    float bv = bias[e * ED_ + ccol];
#pragma unroll
    for (int q = 0; q < 8; ++q)
      C[(size_t)(crow + q) * ED_ + ccol] = u.f[q] + bv;
  }
}

// ---------------- 3) generic fp32 WMMA GEMM, 16x64 tile per wave: C = A @ W[e] + b[e] ----------------
__global__ __launch_bounds__(256) void gemm_wmma_kernel(
    const float* __restrict__ A, const float* __restrict__ W,
    const float* __restrict__ bias, float* __restrict__ C,
    int M, int N, int K, int ldc, int rows_per_expert)
{
  int wave = (blockIdx.x * blockDim.x + threadIdx.x) >> 5;
  int lane = threadIdx.x & 31;
  int ngroups = (N + 63) >> 6;
  int mtiles  = M >> 4;
  if (wave >= mtiles * ngroups) return;
  int mt = wave / ngroups, ng = wave % ngroups;
  int e = (mt << 4) / rows_per_expert;
  const float* Wp = W + (size_t)e * K * N;
  const float* bp = bias + (size_t)e * N;
  int row  = (mt << 4) + (lane & 15);
  int khalf = (lane >> 4) << 1;
  const float* arow = A + (size_t)row * K + khalf;
  int colb = (ng << 6) + (lane & 15);
  bool v0c = (colb      ) < N;
  bool v1c = (colb + 16 ) < N;
  bool v2c = (colb + 32 ) < N;
  bool v3c = (colb + 48 ) < N;
  v8f acc0 = {}, acc1 = {}, acc2 = {}, acc3 = {};
  for (int k0 = 0; k0 < K; k0 += 4) {
    v2f a;
    a.x = arow[k0]; a.y = arow[k0 + 1];
    const float* wp = Wp + (size_t)(k0 + khalf) * N + colb;
    v2f b0, b1, b2, b3;
    if (v0c) { b0.x = wp[0];  b0.y = wp[N];      } else { b0.x = 0.f; b0.y = 0.f; }
    if (v1c) { b1.x = wp[16]; b1.y = wp[N + 16]; } else { b1.x = 0.f; b1.y = 0.f; }
    if (v2c) { b2.x = wp[32]; b2.y = wp[N + 32]; } else { b2.x = 0.f; b2.y = 0.f; }
    if (v3c) { b3.x = wp[48]; b3.y = wp[N + 48]; } else { b3.x = 0.f; b3.y = 0.f; }
    acc0 = __builtin_amdgcn_wmma_f32_16x16x4_f32(false, a, false, b0, (short)0, acc0, false, false);
    acc1 = __builtin_amdgcn_wmma_f32_16x16x4_f32(false, a, false, b1, (short)0, acc1, false, false);
    acc2 = __builtin_amdgcn_wmma_f32_16x16x4_f32(false, a, false, b2, (short)0, acc2, false, false);
    acc3 = __builtin_amdgcn_wmma_f32_16x16x4_f32(false, a, false, b3, (short)0, acc3, false, false);
  }
  int crow = (mt << 4) + ((lane >> 4) << 3);
  union { v8f v; float f[8]; } u;
#pragma unroll
  for (int i = 0; i < 4; ++i) {
    int ccol = (ng << 6) + (i << 4) + (lane & 15);
    if (ccol < N) {
      u.v = (i == 0) ? acc0 : (i == 1) ? acc1 : (i == 2) ? acc2 : acc3;
      float bv = bp[ccol];
#pragma unroll
      for (int q = 0; q < 8; ++q)
        C[(size_t)(crow + q) * ldc + ccol] = u.f[q] + bv;
    }
  }
}

// ---------------- 4) lorentz activation chains (one wave per row) ----------------
// mode 0: logmap0∘projx∘expmap0 applied twice (mid MLP layers)
// mode 1: projx∘expmap0 (final point)
// mode 2: logmap0∘projx∘expmap0 once
__global__ __launch_bounds__(256) void lorentz_act_kernel(
    const float* __restrict__ Y, float* __restrict__ O, int R, int D, int mode)
{
  int wave = (blockIdx.x * blockDim.x + threadIdx.x) >> 5;
  int lane = threadIdx.x & 31;
  if (wave >= R) return;
  const float* y = Y + (size_t)wave * D;
  float* o = O + (size_t)wave * D;
  float s = 0.f;
  for (int j = lane; j < D; j += 32) { float v = y[j]; s += (j == 0) ? 0.f : v * v; }
  float S1 = wsum(s);
  if (mode == 1) {
    float n  = sqrtf(fmaxf(S1, 1e-12f));
    float th = fminf(n, 10.f);
    float f  = sinhf(th) / n;
    float q0 = sqrtf(1.f + f * f * S1);
    for (int j = lane; j < D; j += 32) o[j] = (j == 0) ? q0 : f * y[j];
  } else {
    float g = chain_g(S1);
    if (mode == 0) { float S2 = g * g * S1; g *= chain_g(S2); }
    for (int j = lane; j < D; j += 32) o[j] = (j == 0) ? 0.f : g * y[j];
  }
}

// ---------------- 5) Frechet mean over np points of dim 128 ----------------
__global__ __launch_bounds__(256) void frechet_kernel(
    const float* __restrict__ P, float* __restrict__ O,
    int R, int np, long rowStride, long ptStride)
{
  int wave = (blockIdx.x * blockDim.x + threadIdx.x) >> 5;
  int lane = threadIdx.x & 31;
  if (wave >= R) return;
  const float* base = P + (size_t)wave * rowStride;
  float inv = 1.f / (float)np;

  float m0 = 0, m1 = 0, m2 = 0, m3 = 0;
  for (int p = 0; p < np; ++p) {
    const float* pt = base + (size_t)p * ptStride;
    float a0 = pt[lane], a1 = pt[lane + 32], a2 = pt[lane + 64], a3 = pt[lane + 96];
    float x0 = bcast0(a0);
    float loc = a0 * a0 + a1 * a1 + a2 * a2 + a3 * a3;
    if (lane == 0) loc -= a0 * a0;                  // exclude component 0
    float ss = wsum(loc);
    float d  = acoshf(fmaxf(x0, 1.f + EPS_));
    float c  = d / sqrtf(fmaxf(ss, 1e-12f));
    m0 += c * a0; m1 += c * a1; m2 += c * a2; m3 += c * a3;
  }
  m0 *= inv; m1 *= inv; m2 *= inv; m3 *= inv;
  if (lane == 0) m0 = 0.f;                          // logmap0 comp0 = 0
  float S1 = wsum(m0 * m0 + m1 * m1 + m2 * m2 + m3 * m3);
  float n  = sqrtf(fmaxf(S1, 1e-12f));
  float th = fminf(n, 10.f);
  float f  = sinhf(th) / n;
  float c0 = sqrtf(1.f + f * f * S1);
  float cur0 = f * m0, cur1 = f * m1, cur2 = f * m2, cur3 = f * m3;
  if (lane == 0) cur0 = c0;

  for (int it = 0; it < 10; ++it) {
    float v0 = 0, v1 = 0, v2 = 0, v3 = 0;
    float cx0 = bcast0(cur0);
    for (int p = 0; p < np; ++p) {
      const float* pt = base + (size_t)p * ptStride;
      float a0 = pt[lane], a1 = pt[lane + 32], a2 = pt[lane + 64], a3 = pt[lane + 96];
      float px0  = bcast0(a0);
      float dot  = wsum(cur0 * a0 + cur1 * a1 + cur2 * a2 + cur3 * a3);
      float lin  = dot - 2.f * cx0 * px0;
      float beta = fmaxf(-lin, 1.f + EPS_);
      float d    = acoshf(beta);
      float u0 = a0 - beta * cur0, u1 = a1 - beta * cur1, u2 = a2 - beta * cur2, u3 = a3 - beta * cur3;
      float uz = bcast0(u0);
      float uu = wsum(u0 * u0 + u1 * u1 + u2 * u2 + u3 * u3) - 2.f * uz * uz;
      float w  = d / sqrtf(fmaxf(uu, 1e-12f));
      v0 += w * u0; v1 += w * u1; v2 += w * u2; v3 += w * u3;
    }
    float sc = 0.5f * inv;
    v0 *= sc; v1 *= sc; v2 *= sc; v3 *= sc;
    float vz = bcast0(v0);
    float nn = wsum(v0 * v0 + v1 * v1 + v2 * v2 + v3 * v3) - 2.f * vz * vz;
    float n2 = sqrtf(fmaxf(nn, 1e-12f));
    float t2 = fminf(n2, 10.f);
    float ch = coshf(t2), shv = sinhf(t2) / n2;
    cur0 = ch * cur0 + shv * v0; cur1 = ch * cur1 + shv * v1;
    cur2 = ch * cur2 + shv * v2; cur3 = ch * cur3 + shv * v3;
    float cz  = bcast0(cur0);
    float Ssp = wsum(cur0 * cur0 + cur1 * cur1 + cur2 * cur2 + cur3 * cur3) - cz * cz;
    if (lane == 0) cur0 = sqrtf(1.f + Ssp);
  }
  float* o = O + (size_t)wave * ED_;
  o[lane] = cur0; o[lane + 32] = cur1; o[lane + 64] = cur2; o[lane + 96] = cur3;
}

// ---------------- 6) v_h = logmap0(projx(expmap0(logmap(x_prev, z_last)))) ----------------
__global__ __launch_bounds__(256) void vh_kernel(
    const float* __restrict__ X, const float* __restrict__ Z, float* __restrict__ O)
{
  int wave = (blockIdx.x * blockDim.x + threadIdx.x) >> 5;
  int lane = threadIdx.x & 31;
  if (wave >= B_) return;
  const float* x = X + (size_t)wave * ED_;
  const float* z = Z + (size_t)wave * ED_;
  float x0 = x[lane], x1 = x[lane + 32], x2 = x[lane + 64], x3 = x[lane + 96];
  float z0 = z[lane], z1 = z[lane + 32], z2 = z[lane + 64], z3 = z[lane + 96];
  float xz = bcast0(x0), zz = bcast0(z0);
  float dot  = wsum(x0 * z0 + x1 * z1 + x2 * z2 + x3 * z3);
  float lin  = dot - 2.f * xz * zz;
  float beta = fmaxf(-lin, 1.f + EPS_);
  float d    = acoshf(beta);
  float u0 = z0 - beta * x0, u1 = z1 - beta * x1, u2 = z2 - beta * x2, u3 = z3 - beta * x3;
  float uz = bcast0(u0);
  float uu = wsum(u0 * u0 + u1 * u1 + u2 * u2 + u3 * u3) - 2.f * uz * uz;
  float c  = d / sqrtf(fmaxf(uu, 1e-12f));
  float v0 = c * u0, v1 = c * u1, v2 = c * u2, v3 = c * u3;
  float vz = c * uz;
  float S1 = wsum(v0 * v0 + v1 * v1 + v2 * v2 + v3 * v3) - vz * vz;
  float g  = chain_g(S1);
  float* o = O + (size_t)wave * ED_;
  o[lane] = (lane == 0) ? 0.f : g * v0;
  o[lane + 32] = g * v1; o[lane + 64] = g * v2; o[lane + 96] = g * v3;
}

// ---------------- 7) trajectory update + decoder logmap0 ----------------
__global__ __launch_bounds__(256) void update_kernel(
    const float* __restrict__ Z, const float* __restrict__ V,
    const float* __restrict__ alpha_p, const float* __restrict__ step_p,
    float* __restrict__ ZN, float* __restrict__ UD)
{
  int wave = (blockIdx.x * blockDim.x + threadIdx.x) >> 5;
  int lane = threadIdx.x & 31;
  if (wave >= B_) return;
  const float* z = Z + (size_t)wave * ED_;
  const float* v = V + (size_t)wave * ED_;
  float z0 = z[lane], z1 = z[lane + 32], z2 = z[lane + 64], z3 = z[lane + 96];
  float t0 = v[lane], t1 = v[lane + 32], t2 = v[lane + 64], t3 = v[lane + 96];
  float step = 1.f / (1.f + expf(-step_p[0]));
  float a    = 1.f / (1.f + expf(-alpha_p[0]));
  // z_up = projx(safe_expmap(z_last, step*v_tr))
  float u0 = step * t0, u1 = step * t1, u2 = step * t2, u3 = step * t3;
  float uz = bcast0(u0);
  float uu = wsum(u0 * u0 + u1 * u1 + u2 * u2 + u3 * u3) - 2.f * uz * uz;
  float n  = sqrtf(fmaxf(uu, 1e-12f));
  float th = fminf(n, 10.f);
  float ch = coshf(th), sh = sinhf(th) / n;
  float p0 = ch * z0 + sh * u0, p1 = ch * z1 + sh * u1, p2 = ch * z2 + sh * u2, p3 = ch * z3 + sh * u3;
  float pz = bcast0(p0);
  float Ssp = wsum(p0 * p0 + p1 * p1 + p2 * p2 + p3 * p3) - pz * pz;
  if (lane == 0) p0 = sqrtf(1.f + Ssp);
  // w = (1-a) * logmap(z_last, z_up)
  float zz  = bcast0(z0);
  float pz2 = bcast0(p0);
  float dot  = wsum(z0 * p0 + z1 * p1 + z2 * p2 + z3 * p3);
  float lin  = dot - 2.f * zz * pz2;
  float beta = fmaxf(-lin, 1.f + EPS_);
  float d    = acoshf(beta);
  float q0 = p0 - beta * z0, q1 = p1 - beta * z1, q2 = p2 - beta * z2, q3 = p3 - beta * z3;
  float qz = bcast0(q0);
  float qq = wsum(q0 * q0 + q1 * q1 + q2 * q2 + q3 * q3) - 2.f * qz * qz;
  float c  = (1.f - a) * d / sqrtf(fmaxf(qq, 1e-12f));
  float w0 = c * q0, w1 = c * q1, w2 = c * q2, w3 = c * q3;
  // z_next = projx(safe_expmap(z_last, w))
  float wz  = bcast0(w0);
  float wwl = wsum(w0 * w0 + w1 * w1 + w2 * w2 + w3 * w3) - 2.f * wz * wz;
  float n2  = sqrtf(fmaxf(wwl, 1e-12f));
  float th2 = fminf(n2, 10.f);
  float ch2 = coshf(th2), sh2 = sinhf(th2) / n2;
  float r0 = ch2 * z0 + sh2 * w0, r1 = ch2 * z1 + sh2 * w1, r2 = ch2 * z2 + sh2 * w2, r3 = ch2 * z3 + sh2 * w3;
  float rz   = bcast0(r0);
  float Ssp2 = wsum(r0 * r0 + r1 * r1 + r2 * r2 + r3 * r3) - rz * rz;
  float r0n  = sqrtf(1.f + Ssp2);
  if (lane == 0) r0 = r0n;
  float* zn = ZN + (size_t)wave * ED_;
  zn[lane] = r0; zn[lane + 32] = r1; zn[lane + 64] = r2; zn[lane + 96] = r3;
  // u_dec = logmap0(z_next)
  float d2 = acoshf(fmaxf(r0n, 1.f + EPS_));
  float g  = d2 / sqrtf(fmaxf(Ssp2, 1e-12f));
  float* ud = UD + (size_t)wave * ED_;
  ud[lane] = (lane == 0) ? 0.f : g * r0;
  ud[lane + 32] = g * r1; ud[lane + 64] = g * r2; ud[lane + 96] = g * r3;
}

// ---------------- 8) layernorm + exact gelu, D=256 ----------------
__global__ __launch_bounds__(256) void lngelu_kernel(
    const float* __restrict__ T, const float* __restrict__ G,
    const float* __restrict__ Bv, float* __restrict__ O)
{
  int wave = (blockIdx.x * blockDim.x + threadIdx.x) >> 5;
  int lane = threadIdx.x & 31;
  if (wave >= B_) return;
  const float* t = T + (size_t)wave * HD_;
  float r[8]; float s = 0.f;
#pragma unroll
  for (int q = 0; q < 8; ++q) { r[q] = t[lane + 32 * q]; s += r[q]; }
  float m = wsum(s) / (float)HD_;
  float vs = 0.f;
#pragma unroll
  for (int q = 0; q < 8; ++q) { float dd = r[q] - m; vs += dd * dd; }
  float var = wsum(vs) / (float)HD_;
  float is  = rsqrtf(var + 1e-5f);
  float* o = O + (size_t)wave * HD_;
#pragma unroll
  for (int q = 0; q < 8; ++q) {
    int j = lane + 32 * q;
    float x = (r[q] - m) * is * G[j] + Bv[j];
    o[j] = 0.5f * x * (1.f + erff(x * 0.70710678118654752f));
  }
}

// ---------------- launcher ----------------
extern "C" void kernel_launch(void* const* d_in, const int* in_sizes, int n_in,
                              void* d_out, int out_size, void* d_ws, size_t ws_size,
                              hipStream_t stream)
{
  (void)in_sizes; (void)n_in; (void)out_size; (void)ws_size;
  const float* trend   = (const float*)d_in[0];
  const float* scoarse = (const float*)d_in[1];
  const float* sfine   = (const float*)d_in[2];
  const float* resid   = (const float*)d_in[3];
  const float* enc_W0  = (const float*)d_in[4];
  const float* enc_b0  = (const float*)d_in[5];
  const float* enc_W1  = (const float*)d_in[6];
  const float* enc_b1  = (const float*)d_in[7];
  const float* enc_W2  = (const float*)d_in[8];
  const float* enc_b2  = (const float*)d_in[9];
  const float* eff_sc  = (const float*)d_in[10];
  const float* dyn_W0  = (const float*)d_in[12];
  const float* dyn_b0  = (const float*)d_in[13];
  const float* dyn_W1  = (const float*)d_in[14];
  const float* dyn_b1  = (const float*)d_in[15];
  const float* dyn_W2  = (const float*)d_in[16];
  const float* dyn_b2  = (const float*)d_in[17];
  const float* alpha_p = (const float*)d_in[18];
  const float* step_p  = (const float*)d_in[19];
  const float* dec_W0  = (const float*)d_in[20];
  const float* dec_b0  = (const float*)d_in[21];
  const float* ln_g    = (const float*)d_in[22];
  const float* ln_b    = (const float*)d_in[23];
  const float* dec_W1  = (const float*)d_in[24];
  const float* dec_b1  = (const float*)d_in[25];
  float* out = (float*)d_out;

  float* ws = (float*)d_ws;
  size_t o = 0;
  float* scaleArr = ws + o; o += NROWS_;
  float* shiftArr = ws + o; o += NROWS_;
  float* bufY     = ws + o; o += (size_t)NROWS_ * ED_;
  float* bufH     = ws + o; o += (size_t)NROWS_ * ED_;
  float* pooled   = ws + o; o += (size_t)4 * B_ * ED_;
  float* slots    = ws + o; o += (size_t)(NSTEPS_ + 1) * B_ * ED_;
  float* mlp_in   = ws + o; o += (size_t)B_ * ED_;
  float* b256a    = ws + o; o += (size_t)B_ * HD_;
  float* b256b    = ws + o; o += (size_t)B_ * HD_;
  float* b128     = ws + o; o += (size_t)B_ * ED_;
  float* vtr      = ws + o; o += (size_t)B_ * ED_;
  float* udec     = ws + o; o += (size_t)B_ * ED_;
  float* tbuf     = ws + o; o += (size_t)B_ * HD_;
  float* t2buf    = ws + o; o += (size_t)B_ * HD_;

  dim3 blk(256);
  auto wg = [](long waves) { return dim3((unsigned)((waves + 7) / 8)); };
  auto gemmWaves = [](long M, long N) { return (M >> 4) * ((N + 63) >> 6); };

  // encoder
  stats_kernel<<<wg(NROWS_), blk, 0, stream>>>(trend, scoarse, sfine, resid, eff_sc, scaleArr, shiftArr);
  gemm1_fused_kernel<<<wg((long)(NROWS_ / 16) * (ED_ / 64)), blk, 0, stream>>>(
      trend, scoarse, sfine, resid, scaleArr, shiftArr, enc_W0, enc_b0, bufY);
  lorentz_act_kernel<<<wg(NROWS_), blk, 0, stream>>>(bufY, bufH, NROWS_, ED_, 0);
  gemm_wmma_kernel<<<wg(gemmWaves(NROWS_, ED_)), blk, 0, stream>>>(
      bufH, enc_W1, enc_b1, bufY, NROWS_, ED_, ED_, ED_, RPE_);
  lorentz_act_kernel<<<wg(NROWS_), blk, 0, stream>>>(bufY, bufH, NROWS_, ED_, 0);
  gemm_wmma_kernel<<<wg(gemmWaves(NROWS_, ED_)), blk, 0, stream>>>(
      bufH, enc_W2, enc_b2, bufY, NROWS_, ED_, ED_, ED_, RPE_);
  lorentz_act_kernel<<<wg(NROWS_), blk, 0, stream>>>(bufY, bufH, NROWS_, ED_, 1);  // z points

  // frechet means: over segments, then over the 4 experts
  frechet_kernel<<<wg(4 * B_), blk, 0, stream>>>(bufH, pooled, 4 * B_, NSEG_, (long)NSEG_ * ED_, (long)ED_);
  frechet_kernel<<<wg(B_), blk, 0, stream>>>(pooled, slots, B_, 4, (long)ED_, (long)B_ * ED_);

  // decoder rollout: slots[0]=combined, slots[t+1]=z_next(t)
  for (int t = 0; t < NSTEPS_; ++t) {
    const float* zlast = slots + (size_t)t * B_ * ED_;
    const float* xprev = slots + (size_t)(t == 0 ? 0 : (t - 1)) * B_ * ED_;
    float* znext = slots + (size_t)(t + 1) * B_ * ED_;

    vh_kernel<<<wg(B_), blk, 0, stream>>>(xprev, zlast, mlp_in);
    gemm_wmma_kernel<<<wg(gemmWaves(B_, HD_)), blk, 0, stream>>>(mlp_in, dyn_W0, dyn_b0, b256a, B_, HD_, ED_, HD_, B_);
    lorentz_act_kernel<<<wg(B_), blk, 0, stream>>>(b256a, b256b, B_, HD_, 0);
    gemm_wmma_kernel<<<wg(gemmWaves(B_, HD_)), blk, 0, stream>>>(b256b, dyn_W1, dyn_b1, b256a, B_, HD_, HD_, HD_, B_);
    lorentz_act_kernel<<<wg(B_), blk, 0, stream>>>(b256a, b256b, B_, HD_, 0);
    gemm_wmma_kernel<<<wg(gemmWaves(B_, ED_)), blk, 0, stream>>>(b256b, dyn_W2, dyn_b2, b128, B_, ED_, HD_, ED_, B_);
    lorentz_act_kernel<<<wg(B_), blk, 0, stream>>>(b128, vtr, B_, ED_, 2);   // v_tr = logmap0(mlp out)
    update_kernel<<<wg(B_), blk, 0, stream>>>(zlast, vtr, alpha_p, step_p, znext, udec);
    gemm_wmma_kernel<<<wg(gemmWaves(B_, HD_)), blk, 0, stream>>>(udec, dec_W0, dec_b0, tbuf, B_, HD_, ED_, HD_, B_);
    lngelu_kernel<<<wg(B_), blk, 0, stream>>>(tbuf, ln_g, ln_b, t2buf);
    // final projection straight into d_out: row stride 96*21, column offset t*504
    gemm_wmma_kernel<<<wg(gemmWaves(B_, INP_)), blk, 0, stream>>>(
        t2buf, dec_W1, dec_b1, out + (size_t)t * INP_, B_, INP_, HD_, 96 * NF_, B_);
  }
}